// SymmetricCrossAttention_6004364280543
// MI455X (gfx1250) — compile-verified
//
#include <hip/hip_runtime.h>

typedef __attribute__((ext_vector_type(2))) float v2f;
typedef __attribute__((ext_vector_type(4))) float v4f;
typedef __attribute__((ext_vector_type(8))) float v8f;

#define NCH   64
#define NPIX  65536
#define NB    4
#define LSTR  68   // LDS row stride: 4-float aligned, bank stride 4 -> conflict-free

static __device__ __forceinline__ v8f wmma4(v2f a, v2f b, v8f c) {
  // V_WMMA_F32_16X16X4_F32 : D = A(16x4 f32) * B(4x16 f32) + C, exact fp32
  return __builtin_amdgcn_wmma_f32_16x16x4_f32(false, a, false, b, (short)0, c,
                                               false, false);
}

// ---------------------------------------------------------------------------
// Pass A: per batch, G = X1 * X2^T (64x64) and s1 = sum_n X1[:,n], s2 likewise.
// grid (64, B), block 256 (8 waves); each block reduces a 1024-pixel chunk.
// ---------------------------------------------------------------------------
__global__ __launch_bounds__(256) void gram_kernel(
    const float* __restrict__ x1, const float* __restrict__ x2,
    float* __restrict__ G, float* __restrict__ s1, float* __restrict__ s2) {
  __shared__ float sX1[NCH * LSTR];
  __shared__ float sX2[NCH * LSTR];

  const int b    = blockIdx.y;
  const int t    = threadIdx.x;
  const int wv   = t >> 5;
  const int lane = t & 31;
  const int l16  = lane & 15;
  const int h    = lane >> 4;
  const int mi   = wv >> 1;          // waves 2w,2w+1 share the A row-group
  const int ni0  = (2 * wv) & 3;
  const int ni1  = (2 * wv + 1) & 3;

  const float* x1b = x1 + (size_t)b * NCH * NPIX;
  const float* x2b = x2 + (size_t)b * NCH * NPIX;

  v8f acc0 = {};
  v8f acc1 = {};
  float srow = 0.0f;

  const int chunk0 = blockIdx.x * 1024;
  for (int it = 0; it < 16; ++it) {
    const int p0  = chunk0 + it * 64;
    const int px4 = (t & 15) * 4;
    #pragma unroll
    for (int r = 0; r < 4; ++r) {     // coalesced float4 stage, [ch][px] layout
      const int ch = (t >> 4) + r * 16;
      const size_t g = (size_t)ch * NPIX + p0 + px4;
      *(v4f*)&sX1[ch * LSTR + px4] = *(const v4f*)&x1b[g];
      *(v4f*)&sX2[ch * LSTR + px4] = *(const v4f*)&x2b[g];
    }
    __syncthreads();

    if (t < 64) {                     // per-channel pixel sums (wave-uniform)
      const float* row = &sX1[t * LSTR];
      #pragma unroll
      for (int p = 0; p < 64; p += 4) {
        v4f v = *(const v4f*)&row[p];
        srow += v.x + v.y + v.z + v.w;
      }
    } else if (t < 128) {
      const float* row = &sX2[(t - 64) * LSTR];
      #pragma unroll
      for (int p = 0; p < 64; p += 4) {
        v4f v = *(const v4f*)&row[p];
        srow += v.x + v.y + v.z + v.w;
      }
    }

    #pragma unroll
    for (int kk = 0; kk < 16; ++kk) { // K = 64 pixels in steps of 4
      const int k = kk * 4 + 2 * h;   // fp32 frag: lanes<16 K=0,1; lanes>=16 K=2,3
      v2f a  = *(const v2f*)&sX1[(mi  * 16 + l16) * LSTR + k];
      v2f b0 = *(const v2f*)&sX2[(ni0 * 16 + l16) * LSTR + k];
      v2f b1 = *(const v2f*)&sX2[(ni1 * 16 + l16) * LSTR + k];
      acc0 = wmma4(a, b0, acc0);
      acc1 = wmma4(a, b1, acc1);
    }
    __syncthreads();
  }

  float* Gb = G + b * NCH * NCH;
  #pragma unroll
  for (int r = 0; r < 8; ++r) {       // D layout: VGPR r -> M = r + 8*h, N = l16
    const int m = mi * 16 + r + 8 * h;
    atomicAdd(&Gb[m * NCH + ni0 * 16 + l16], acc0[r]);
    atomicAdd(&Gb[m * NCH + ni1 * 16 + l16], acc1[r]);
  }
  if (t < 64)        atomicAdd(&s1[b * NCH + t], srow);
  else if (t < 128)  atomicAdd(&s2[b * NCH + (t - 64)], srow);
}

// ---------------------------------------------------------------------------
// Pass B (tiny): logits -> softmax -> fold projections into M1/M2, c1/c2.
// grid B, block 64; thread c owns row c of every 64x64 matrix.
// ---------------------------------------------------------------------------
__global__ __launch_bounds__(64) void attn_fold_kernel(
    const float* __restrict__ G, const float* __restrict__ s1v,
    const float* __restrict__ s2v,
    const float* wq1, const float* bq1, const float* wk2, const float* bk2,
    const float* wv2, const float* bv2, const float* wq2, const float* bq2,
    const float* wk1, const float* bk1, const float* wv1, const float* bv1,
    const float* wo1, const float* bo1, const float* wo2, const float* bo2,
    float* __restrict__ M1, float* __restrict__ c1,
    float* __restrict__ M2, float* __restrict__ c2) {
  __shared__ float sG[64 * 65], sA[64 * 65], sAt[64 * 65], sP[64 * 65];
  __shared__ float sVq[64], sVk[64], sVt[64], sS1[64], sS2[64];

  const int c = threadIdx.x;
  const int b = blockIdx.x;
  const float scale = 0.125f;  // 64^-0.5

  for (int j = 0; j < 64; ++j) sG[c * 65 + j] = G[b * 4096 + c * 64 + j];
  sS1[c] = s1v[b * 64 + c];
  sS2[c] = s2v[b * 64 + c];
  __syncthreads();

  for (int path = 0; path < 2; ++path) {
    const float *Wq, *bq, *Wk, *bk, *Wv, *bv, *Wo, *bo, *sQ, *sK;
    float *Mo, *co;
    if (path == 0) { Wq=wq1; bq=bq1; Wk=wk2; bk=bk2; Wv=wv2; bv=bv2;
                     Wo=wo1; bo=bo1; sQ=sS1; sK=sS2; Mo=M1; co=c1; }
    else           { Wq=wq2; bq=bq2; Wk=wk1; bk=bk1; Wv=wv1; bv=bv1;
                     Wo=wo2; bo=bo2; sQ=sS2; sK=sS1; Mo=M2; co=c2; }

    for (int j = 0; j < 64; ++j) {             // A = Wq * (G or G^T)
      float acc = 0.f;
      for (int i = 0; i < 64; ++i)
        acc += Wq[c * 64 + i] * (path ? sG[j * 65 + i] : sG[i * 65 + j]);
      sA[c * 65 + j] = acc;
    }
    {
      float u = 0.f, w = 0.f;
      for (int i = 0; i < 64; ++i) {
        u += Wq[c * 64 + i] * sQ[i];
        w += Wk[c * 64 + i] * sK[i];
      }
      sVq[c] = u; sVk[c] = w;
    }
    __syncthreads();

    float mx = -3.4e38f;                       // logits + rowwise softmax
    for (int d = 0; d < 64; ++d) {
      float acc = 0.f;
      for (int j = 0; j < 64; ++j) acc += sA[c * 65 + j] * Wk[d * 64 + j];
      acc += sVq[c] * bk[d] + bq[c] * sVk[d] + (float)NPIX * bq[c] * bk[d];
      acc *= scale;
      sAt[c * 65 + d] = acc;
      mx = fmaxf(mx, acc);
    }
    float sum = 0.f;
    for (int d = 0; d < 64; ++d) {
      float e = __expf(sAt[c * 65 + d] - mx);
      sAt[c * 65 + d] = e; sum += e;
    }
    const float inv = 1.0f / sum;
    for (int d = 0; d < 64; ++d) sAt[c * 65 + d] *= inv;
    __syncthreads();

    {
      float tacc = 0.f;                        // P = attn*Wv ; t = attn*bv
      for (int j = 0; j < 64; ++j) tacc += sAt[c * 65 + j] * bv[j];
      sVt[c] = tacc;
      for (int d = 0; d < 64; ++d) {
        float acc = 0.f;
        for (int j = 0; j < 64; ++j) acc += sAt[c * 65 + j] * Wv[j * 64 + d];
        sP[c * 65 + d] = acc;
      }
    }
    __syncthreads();

    {
      float ca = 0.f;                          // M = Wo*P ; c = Wo*t + bo
      for (int i = 0; i < 64; ++i) ca += Wo[c * 64 + i] * sVt[i];
      co[b * 64 + c] = ca + bo[c];
      for (int d = 0; d < 64; ++d) {
        float acc = 0.f;
        for (int i = 0; i < 64; ++i) acc += Wo[c * 64 + i] * sP[i * 65 + d];
        Mo[b * 4096 + c * 64 + d] = acc;
      }
    }
    __syncthreads();
  }
}

// ---------------------------------------------------------------------------
// Pass C: out1 = M1*X2 + c1 + X1 ; out2 = M2*X1 + c2 + X2 (fused, streaming).
// grid (256, B), block 256; 256-pixel tile per block, 4 x 64-px WMMA subtiles.
// ---------------------------------------------------------------------------
__global__ __launch_bounds__(256) void apply_kernel(
    const float* __restrict__ x1, const float* __restrict__ x2,
    const float* __restrict__ M1, const float* __restrict__ c1,
    const float* __restrict__ M2, const float* __restrict__ c2,
    float* __restrict__ out1, float* __restrict__ out2) {
  __shared__ float sM1[NCH * LSTR], sM2[NCH * LSTR];
  __shared__ float sX1t[NCH * LSTR], sX2t[NCH * LSTR];   // transposed [px][ch]
  __shared__ float sOut[NCH * LSTR];
  __shared__ float sC1[NCH], sC2[NCH];

  const int b    = blockIdx.y;
  const int t    = threadIdx.x;
  const int wv   = t >> 5;
  const int lane = t & 31;
  const int l16  = lane & 15;
  const int h    = lane >> 4;
  const int mi   = wv >> 1;
  const int ni0  = (2 * wv) & 3;
  const int ni1  = (2 * wv + 1) & 3;

  const float* x1b = x1 + (size_t)b * NCH * NPIX;
  const float* x2b = x2 + (size_t)b * NCH * NPIX;
  float* o1b = out1 + (size_t)b * NCH * NPIX;
  float* o2b = out2 + (size_t)b * NCH * NPIX;

  #pragma unroll
  for (int r = 0; r < 4; ++r) {                 // stage M1/M2 (float4, aligned)
    const int i4 = t * 4 + r * 1024;
    const int row = i4 >> 6, col = i4 & 63;
    *(v4f*)&sM1[row * LSTR + col] = *(const v4f*)&M1[b * 4096 + i4];
    *(v4f*)&sM2[row * LSTR + col] = *(const v4f*)&M2[b * 4096 + i4];
  }
  if (t < 64) { sC1[t] = c1[b * 64 + t]; sC2[t] = c2[b * 64 + t]; }
  __syncthreads();

  const int px4 = (t & 15) * 4;
  for (int sub = 0; sub < 4; ++sub) {
    const int p0 = blockIdx.x * 256 + sub * 64;
    #pragma unroll
    for (int r = 0; r < 4; ++r) {               // coalesced load, transposed store
      const int ch = (t >> 4) + r * 16;
      const size_t g = (size_t)ch * NPIX + p0 + px4;
      v4f v1 = *(const v4f*)&x1b[g];
      v4f v2 = *(const v4f*)&x2b[g];
      sX1t[(px4 + 0) * LSTR + ch] = v1.x; sX1t[(px4 + 1) * LSTR + ch] = v1.y;
      sX1t[(px4 + 2) * LSTR + ch] = v1.z; sX1t[(px4 + 3) * LSTR + ch] = v1.w;
      sX2t[(px4 + 0) * LSTR + ch] = v2.x; sX2t[(px4 + 1) * LSTR + ch] = v2.y;
      sX2t[(px4 + 2) * LSTR + ch] = v2.z; sX2t[(px4 + 3) * LSTR + ch] = v2.w;
    }
    __syncthreads();

    v8f a10 = {}, a11 = {}, a20 = {}, a21 = {};
    #pragma unroll
    for (int kk = 0; kk < 16; ++kk) {           // K = 64 input channels
      const int k = kk * 4 + 2 * h;
      v2f am1 = *(const v2f*)&sM1 [(mi  * 16 + l16) * LSTR + k];
      v2f am2 = *(const v2f*)&sM2 [(mi  * 16 + l16) * LSTR + k];
      v2f b20 = *(const v2f*)&sX2t[(ni0 * 16 + l16) * LSTR + k];
      v2f b21 = *(const v2f*)&sX2t[(ni1 * 16 + l16) * LSTR + k];
      v2f b10 = *(const v2f*)&sX1t[(ni0 * 16 + l16) * LSTR + k];
      v2f b11 = *(const v2f*)&sX1t[(ni1 * 16 + l16) * LSTR + k];
      a10 = wmma4(am1, b20, a10);
      a11 = wmma4(am1, b21, a11);
      a20 = wmma4(am2, b10, a20);
      a21 = wmma4(am2, b11, a21);
    }

    #pragma unroll
    for (int r = 0; r < 8; ++r) {               // stage out1 tile into LDS
      const int m = mi * 16 + r + 8 * h;
      sOut[m * LSTR + ni0 * 16 + l16] = a10[r];
      sOut[m * LSTR + ni1 * 16 + l16] = a11[r];
    }
    __syncthreads();
    #pragma unroll
    for (int r = 0; r < 4; ++r) {               // coalesced store + bias + residual
      const int ch = (t >> 4) + r * 16;
      const float cc = sC1[ch];
      v4f v;
      v.x = sOut[ch * LSTR + px4 + 0] + cc + sX1t[(px4 + 0) * LSTR + ch];
      v.y = sOut[ch * LSTR + px4 + 1] + cc + sX1t[(px4 + 1) * LSTR + ch];
      v.z = sOut[ch * LSTR + px4 + 2] + cc + sX1t[(px4 + 2) * LSTR + ch];
      v.w = sOut[ch * LSTR + px4 + 3] + cc + sX1t[(px4 + 3) * LSTR + ch];
      *(v4f*)&o1b[(size_t)ch * NPIX + p0 + px4] = v;
    }
    __syncthreads();

    #pragma unroll
    for (int r = 0; r < 8; ++r) {               // stage out2 tile
      const int m = mi * 16 + r + 8 * h;
      sOut[m * LSTR + ni0 * 16 + l16] = a20[r];
      sOut[m * LSTR + ni1 * 16 + l16] = a21[r];
    }
    __syncthreads();
    #pragma unroll
    for (int r = 0; r < 4; ++r) {
      const int ch = (t >> 4) + r * 16;
      const float cc = sC2[ch];
      v4f v;
      v.x = sOut[ch * LSTR + px4 + 0] + cc + sX2t[(px4 + 0) * LSTR + ch];
      v.y = sOut[ch * LSTR + px4 + 1] + cc + sX2t[(px4 + 1) * LSTR + ch];
      v.z = sOut[ch * LSTR + px4 + 2] + cc + sX2t[(px4 + 2) * LSTR + ch];
      v.w = sOut[ch * LSTR + px4 + 3] + cc + sX2t[(px4 + 3) * LSTR + ch];
      *(v4f*)&o2b[(size_t)ch * NPIX + p0 + px4] = v;
    }
    __syncthreads();
  }
}

extern "C" void kernel_launch(void* const* d_in, const int* in_sizes, int n_in,
                              void* d_out, int out_size, void* d_ws, size_t ws_size,
                              hipStream_t stream) {
  const float* x1  = (const float*)d_in[0];
  const float* x2  = (const float*)d_in[1];
  const float* wq1 = (const float*)d_in[2];  const float* bq1 = (const float*)d_in[3];
  const float* wk2 = (const float*)d_in[4];  const float* bk2 = (const float*)d_in[5];
  const float* wv2 = (const float*)d_in[6];  const float* bv2 = (const float*)d_in[7];
  const float* wq2 = (const float*)d_in[8];  const float* bq2 = (const float*)d_in[9];
  const float* wk1 = (const float*)d_in[10]; const float* bk1 = (const float*)d_in[11];
  const float* wv1 = (const float*)d_in[12]; const float* bv1 = (const float*)d_in[13];
  const float* wo1 = (const float*)d_in[14]; const float* bo1 = (const float*)d_in[15];
  const float* wo2 = (const float*)d_in[16]; const float* bo2 = (const float*)d_in[17];

  float* ws = (float*)d_ws;
  float* G  = ws;                 // NB*64*64
  float* s1 = ws + 16384;         // NB*64
  float* s2 = ws + 16640;         // NB*64
  float* M1 = ws + 16896;         // NB*64*64
  float* M2 = ws + 33280;         // NB*64*64
  float* c1 = ws + 49664;         // NB*64
  float* c2 = ws + 49920;         // NB*64

  float* out1 = (float*)d_out;
  float* out2 = out1 + (size_t)NB * NCH * NPIX;

  hipMemsetAsync(ws, 0, (16384 + 512) * sizeof(float), stream);

  gram_kernel<<<dim3(64, NB), 256, 0, stream>>>(x1, x2, G, s1, s2);
  attn_fold_kernel<<<dim3(NB), 64, 0, stream>>>(
      G, s1, s2, wq1, bq1, wk2, bk2, wv2, bv2, wq2, bq2,
      wk1, bk1, wv1, bv1, wo1, bo1, wo2, bo2, M1, c1, M2, c2);
  apply_kernel<<<dim3(256, NB), 256, 0, stream>>>(x1, x2, M1, c1, M2, c2,
                                                  out1, out2);
}